// GCNEncoder_4269197492516
// MI455X (gfx1250) — compile-verified
//
#include <hip/hip_runtime.h>
#include <hip/hip_bf16.h>

typedef __attribute__((ext_vector_type(2))) float v2f;
typedef __attribute__((ext_vector_type(8))) float v8f;

#define D_FEAT 128

// ---------------------------------------------------------------------------
// Zero a float buffer (grid-stride).
// ---------------------------------------------------------------------------
__global__ void sage_zero_kernel(float* __restrict__ p, long long n) {
    long long i = (long long)blockIdx.x * blockDim.x + threadIdx.x;
    long long stride = (long long)gridDim.x * blockDim.x;
    for (; i < n; i += stride) p[i] = 0.0f;
}

// ---------------------------------------------------------------------------
// Edge scatter: agg[dst] += feat[src] (128 f32 per edge), optional degree.
// 32 lanes per edge, 4 floats per lane (float4 coalesced gather).
// ---------------------------------------------------------------------------
__global__ void sage_scatter_kernel(const float* __restrict__ feat,
                                    const long long* __restrict__ src,
                                    const long long* __restrict__ dst,
                                    float* __restrict__ agg,
                                    float* __restrict__ deg,
                                    long long n_edges,
                                    int accum_deg) {
    const int lane = threadIdx.x & 31;
    const long long edge = (long long)blockIdx.x * (blockDim.x >> 5)
                         + (threadIdx.x >> 5);
    if (edge >= n_edges) return;

    const long long s = src[edge];
    const long long d = dst[edge];

    const float4 v = *(const float4*)(feat + s * D_FEAT + lane * 4);
    float* a = agg + d * D_FEAT + lane * 4;
    atomicAdd(a + 0, v.x);
    atomicAdd(a + 1, v.y);
    atomicAdd(a + 2, v.z);
    atomicAdd(a + 3, v.w);
    if (accum_deg && lane == 0) atomicAdd(deg + d, 1.0f);
}

// ---------------------------------------------------------------------------
// In-place mean: agg[n][k] /= max(deg[n], 1)
// ---------------------------------------------------------------------------
__global__ void sage_mean_kernel(float* __restrict__ agg,
                                 const float* __restrict__ deg,
                                 long long n_elems) {
    long long i = (long long)blockIdx.x * blockDim.x + threadIdx.x;
    if (i >= n_elems) return;
    const float d = deg[i >> 7];                  // i / 128
    const float inv = 1.0f / fmaxf(d, 1.0f);
    agg[i] *= inv;
}

// ---------------------------------------------------------------------------
// Fused SAGE linear layer via V_WMMA_F32_16X16X4_F32 (exact f32):
//   out[m, o] = sum_k mean[m,k]*Wl[o,k] + sum_k x[m,k]*Wr[o,k] + b[o]
// One wave per 16-node x 16-output tile; K-loop in steps of 4, two WMMAs
// per step (Wl and Wr paths) into a shared f32 accumulator.
//
// VGPR layouts (ISA 7.12.2, wave32):
//   A 16x4 f32 : lanes 0-15 -> M=lane, K=koff..koff+1 with koff = (lane>=16)*2
//   B  4x16 f32: lanes 0-15 -> N=lane (col), same K split; B = W^T so the
//                column o0+col of B is the row o0+col of W (row-major read)
//   C/D 16x16  : col = lane&15, VGPR r -> M = r + (lane>=16)*8
// Tile index is wave-uniform -> EXEC all-ones for every WMMA.
// ---------------------------------------------------------------------------
__global__ void sage_gemm_kernel(const float* __restrict__ mean,
                                 const float* __restrict__ xin,
                                 const float* __restrict__ Wl,
                                 const float* __restrict__ Wr,
                                 const float* __restrict__ bias,
                                 float* __restrict__ out,
                                 int n_nodes) {
    const int lane = threadIdx.x & 31;
    const int wave = threadIdx.x >> 5;
    const int n_otiles = D_FEAT / 16;                        // 8
    const int tile = blockIdx.x * (blockDim.x >> 5) + wave;  // wave-uniform
    const int m0 = (tile / n_otiles) * 16;
    const int o0 = (tile % n_otiles) * 16;
    if (m0 >= n_nodes) return;   // uniform per wave: EXEC stays all-ones

    const int sub  = lane & 15;          // A row / B col / C col
    const int koff = (lane >> 4) << 1;   // 0 for lanes 0-15, 2 for 16-31

    const float* mrow = mean + (size_t)(m0 + sub) * D_FEAT + koff;
    const float* xrow = xin  + (size_t)(m0 + sub) * D_FEAT + koff;
    const float* wlr  = Wl   + (size_t)(o0 + sub) * D_FEAT + koff;
    const float* wrr  = Wr   + (size_t)(o0 + sub) * D_FEAT + koff;

    v8f acc = {};
#pragma unroll
    for (int k0 = 0; k0 < D_FEAT; k0 += 4) {
        v2f a0 = *(const v2f*)(mrow + k0);
        v2f b0 = *(const v2f*)(wlr + k0);
        acc = __builtin_amdgcn_wmma_f32_16x16x4_f32(
            false, a0, false, b0, (short)0, acc, false, false);
        v2f a1 = *(const v2f*)(xrow + k0);
        v2f b1 = *(const v2f*)(wrr + k0);
        acc = __builtin_amdgcn_wmma_f32_16x16x4_f32(
            false, a1, false, b1, (short)0, acc, false, false);
    }

    const float bb = bias[o0 + sub];
    const int mbase = m0 + ((lane >> 4) << 3);   // +8 rows for upper half-wave
    float* orow = out + (size_t)mbase * D_FEAT + o0 + sub;
#pragma unroll
    for (int r = 0; r < 8; ++r) {
        orow[(size_t)r * D_FEAT] = acc[r] + bb;
    }
}

// ---------------------------------------------------------------------------
// Host-side launcher. Workspace layout (floats): agg[N*128] | deg[N] | h[N*128]
// ---------------------------------------------------------------------------
extern "C" void kernel_launch(void* const* d_in, const int* in_sizes, int n_in,
                              void* d_out, int out_size, void* d_ws, size_t ws_size,
                              hipStream_t stream) {
    const float*      x   = (const float*)d_in[0];
    const long long*  ei  = (const long long*)d_in[1];   // int64 [2, E]
    const float*      Wl1 = (const float*)d_in[2];
    const float*      Wr1 = (const float*)d_in[3];
    const float*      b1  = (const float*)d_in[4];
    const float*      Wl2 = (const float*)d_in[5];
    const float*      Wr2 = (const float*)d_in[6];
    const float*      b2  = (const float*)d_in[7];
    float*            out = (float*)d_out;

    const long long N = in_sizes[0] / D_FEAT;   // 100000
    const long long E = in_sizes[1] / 2;        // 1600000

    float* agg = (float*)d_ws;                    // N*128
    float* deg = agg + N * D_FEAT;                // N   (contiguous after agg)
    float* h   = deg + N;                         // N*128

    const long long* src = ei;
    const long long* dst = ei + E;

    const int ZB = 256;
    const long long aggdeg = N * D_FEAT + N;
    dim3 zgrid((unsigned)((aggdeg + ZB - 1) / ZB));

    const int edges_per_block = 256 / 32;                         // 8
    dim3 sgrid((unsigned)((E + edges_per_block - 1) / edges_per_block));

    dim3 mgrid((unsigned)((N * D_FEAT + ZB - 1) / ZB));

    const long long n_tiles = (N / 16) * (D_FEAT / 16);           // 50000
    dim3 ggrid((unsigned)((n_tiles + 7) / 8));                    // 8 waves/block

    // ---- Layer 1 ----
    sage_zero_kernel<<<zgrid, ZB, 0, stream>>>(agg, aggdeg);      // agg + deg
    sage_scatter_kernel<<<sgrid, 256, 0, stream>>>(x, src, dst, agg, deg, E, 1);
    sage_mean_kernel<<<mgrid, ZB, 0, stream>>>(agg, deg, N * D_FEAT);
    sage_gemm_kernel<<<ggrid, 256, 0, stream>>>(agg, x, Wl1, Wr1, b1, h, (int)N);

    // ---- Layer 2 (deg is reused; only agg re-zeroed) ----
    dim3 zgrid2((unsigned)((N * D_FEAT + ZB - 1) / ZB));
    sage_zero_kernel<<<zgrid2, ZB, 0, stream>>>(agg, N * D_FEAT);
    sage_scatter_kernel<<<sgrid, 256, 0, stream>>>(h, src, dst, agg, deg, E, 0);
    sage_mean_kernel<<<mgrid, ZB, 0, stream>>>(agg, deg, N * D_FEAT);
    sage_gemm_kernel<<<ggrid, 256, 0, stream>>>(agg, h, Wl2, Wr2, b2, out, (int)N);
}